// RotationGate_39737037423400
// MI455X (gfx1250) — compile-verified
//
#include <hip/hip_runtime.h>

typedef __attribute__((ext_vector_type(2))) float v2f;
typedef __attribute__((ext_vector_type(4))) float v4f;
typedef __attribute__((ext_vector_type(8))) float v8f;

// State: (2,)^20 x 32 f32. Qubit 9 -> bit weight 1024 in q-space.
// q = hi*2048 + bit*1024 + lo,  hi in [0,512), lo in [0,1024).
// flat float index = q*32 + b.
// Block tile: one hi, 16 lo values, all 32 batches, both bits.
//   in:  4 contiguous blocks of 512 floats (R0,R1,I0,I1) = 8 KiB
//   out: 2 contiguous blocks of 1024 floats (complex interleaved) = 8 KiB

__global__ __launch_bounds__(256) void rx_gate_wmma_kernel(
    const float* __restrict__ theta,
    const float* __restrict__ sre,
    const float* __restrict__ sim,
    float* __restrict__ out)
{
    // float idx: [0,512) R0  [512,1024) R1  [1024,1536) I0  [1536,2048) I1
    //            [2048,3072) OUT bit0  [3072,4096) OUT bit1
    __shared__ float lds[4096];

    const int tid = threadIdx.x;
    const int blk = blockIdx.x;
    const int hi  = blk >> 6;
    const int lo0 = (blk & 63) << 4;
    const long long base0 = ((long long)hi * 2048 + lo0) * 32; // float index of (hi,bit0,lo0,b0)

    // ---------------- async stage: global -> LDS (coalesced b128) --------------
    {
        const unsigned o    = (unsigned)tid;                       // 256 b128 ops per array
        const unsigned voff = ((o & 127u) << 4) + ((o >> 7) * 131072u); // bit1 = +32768 floats
        const unsigned ldsR = ((o >> 7) * 2048u) + ((o & 127u) << 4);
        const unsigned ldsBase = (unsigned)(uintptr_t)&lds[0];     // low 32 bits = LDS byte offset
        const unsigned long long baseR = (unsigned long long)(uintptr_t)(sre + base0);
        const unsigned long long baseI = (unsigned long long)(uintptr_t)(sim + base0);
        asm volatile("global_load_async_to_lds_b128 %0, %1, %2"
                     :: "v"(ldsBase + ldsR), "v"(voff), "s"(baseR) : "memory");
        asm volatile("global_load_async_to_lds_b128 %0, %1, %2"
                     :: "v"(ldsBase + 4096u + ldsR), "v"(voff), "s"(baseI) : "memory");
        asm volatile("s_wait_asynccnt 0x0" ::: "memory");
    }
    __syncthreads();

    // ---------------- per-wave WMMA: D(16x16) = sum_j A_j(16x4) B_j(4x16) ------
    const int lane   = tid & 31;
    const int g      = tid >> 5;      // batch group: batches 4g..4g+3
    const int n      = lane & 15;     // A: row r = n ; B: column (site lo) = n
    const int hiHalf = lane >> 4;     // 0: A cols 0,1 / B rows 0,1 ; 1: A cols 2,3 / B rows 2,3
    const int jj     = n >> 2;        // batch-in-group owning row r
    const int comp   = n & 3;         // output component of row r

    const float th = theta[4 * g + jj];
    const float c  = cosf(0.5f * th);
    const float s  = sinf(0.5f * th);

    // M = [[c,0,0,s],[0,c,-s,0],[0,s,c,0],[-s,0,0,c]] acting on (a0,b0,a1,b1)
    const float avx0 = (comp == 0) ? c  : ((comp == 3) ? -s : 0.0f); // col0
    const float avy0 = (comp == 1) ? c  : ((comp == 2) ?  s : 0.0f); // col1
    const float avx1 = (comp == 1) ? -s : ((comp == 2) ?  c : 0.0f); // col2
    const float avy1 = (comp == 0) ?  s : ((comp == 3) ?  c : 0.0f); // col3
    const v2f av = { hiHalf ? avx1 : avx0, hiHalf ? avy1 : avy0 };
    const v2f z2 = { 0.0f, 0.0f };

    // B rows for this half-wave: (a0,b0) or (a1,b1); 4 batches contiguous -> b128
    const int rbase = hiHalf * 512 + n * 32 + 4 * g;
    const v4f re4 = *(const v4f*)&lds[rbase];          // real: a-components
    const v4f im4 = *(const v4f*)&lds[1024 + rbase];   // imag: b-components

    v8f acc = {};
    {
        const v2f a0 = (jj == 0) ? av : z2; const v2f b0 = { re4.x, im4.x };
        acc = __builtin_amdgcn_wmma_f32_16x16x4_f32(false, a0, false, b0, (short)0, acc, false, false);
        const v2f a1 = (jj == 1) ? av : z2; const v2f b1 = { re4.y, im4.y };
        acc = __builtin_amdgcn_wmma_f32_16x16x4_f32(false, a1, false, b1, (short)0, acc, false, false);
        const v2f a2 = (jj == 2) ? av : z2; const v2f b2 = { re4.z, im4.z };
        acc = __builtin_amdgcn_wmma_f32_16x16x4_f32(false, a2, false, b2, (short)0, acc, false, false);
        const v2f a3 = (jj == 3) ? av : z2; const v2f b3 = { re4.w, im4.w };
        acc = __builtin_amdgcn_wmma_f32_16x16x4_f32(false, a3, false, b3, (short)0, acc, false, false);
    }

    // D rows r = v + 8*hiHalf: batch j=r>>2, comp'=r&3. Pairs (2p,2p+1) are (re,im).
#pragma unroll
    for (int p = 0; p < 4; ++p) {
        const int j   = (p >> 1) + 2 * hiHalf;
        const int b   = 4 * g + j;
        const int bit = p & 1;
        const int idx = 2048 + bit * 1024 + (n * 32 + b) * 2;
        *(v2f*)&lds[idx] = (v2f){ acc[2 * p], acc[2 * p + 1] };
    }
    __syncthreads();

    // ---------------- LDS -> global, coalesced b128, interleaved complex -------
#pragma unroll
    for (int t = 0; t < 2; ++t) {
        const int o   = tid + t * 256;     // 512 b128 ops total
        const int bit = o >> 8;
        const int i4  = o & 255;
        const v4f val = *(const v4f*)&lds[2048 + bit * 1024 + i4 * 4];
        *(v4f*)(out + base0 * 2 + (long long)bit * 65536 + (long long)i4 * 4) = val;
    }
}

extern "C" void kernel_launch(void* const* d_in, const int* in_sizes, int n_in,
                              void* d_out, int out_size, void* d_ws, size_t ws_size,
                              hipStream_t stream) {
    (void)in_sizes; (void)n_in; (void)out_size; (void)d_ws; (void)ws_size;
    const float* theta = (const float*)d_in[0];
    const float* sre   = (const float*)d_in[1];
    const float* sim   = (const float*)d_in[2];
    float* out         = (float*)d_out;
    // 512 hi-slices * 64 lo-chunks
    rx_gate_wmma_kernel<<<dim3(32768), dim3(256), 0, stream>>>(theta, sre, sim, out);
}